// AnchorTarget_76682346103245
// MI455X (gfx1250) — compile-verified
//
#include <hip/hip_runtime.h>

#define HH 128
#define WW 128
#define AA 9
#define NGT 200
#define NANCH (HH*WW*AA)
#define ASTRIDE 8
#define TPB 256
#define NBLK (NANCH/TPB)

// fkey(-1.0f) = ~0xBF800000u
#define KEY_NEG1 0x407FFFFFu

// Base anchors from _generate_base_anchors(16,(0.5,1,2),(8,16,32)), ratio-major.
__constant__ float BA[AA*4] = {
   -84.f,  -40.f,  99.f,  55.f,
  -176.f,  -88.f, 191.f, 103.f,
  -360.f, -184.f, 375.f, 199.f,
   -56.f,  -56.f,  71.f,  71.f,
  -120.f, -120.f, 135.f, 135.f,
  -248.f, -248.f, 263.f, 263.f,
   -36.f,  -80.f,  51.f,  95.f,
   -80.f, -168.f,  95.f, 183.f,
  -168.f, -344.f, 183.f, 359.f
};

typedef unsigned int u32;
typedef u32 v4u __attribute__((ext_vector_type(4)));
typedef int  v8i __attribute__((ext_vector_type(8)));
typedef int  v4i __attribute__((ext_vector_type(4)));

// ---- Tensor Data Mover: DMA the 200x4 f32 gt boxes (3200 B) global -> LDS ----
// D# per CDNA5 ISA: 1D tile of 800 f32 elements.
// This toolchain's builtin is the 6-arg form:
//   (uint32x4 g0, int32x8 g1, int32x4 g2, int32x4 g3, int32x8 g4, i32 cpol)
__device__ __forceinline__ void tdm_load_gt(const float* __restrict__ g, float* lds)
{
  unsigned long long ga = (unsigned long long)(const void*)g;
  u32 lds_off = (u32)(unsigned long long)(void*)lds;  // generic->LDS: low 32 bits are LDS byte offset

  v4u g0;
  g0[0] = 1u;                                   // count=1, is_restore=0, gather=0
  g0[1] = lds_off;                              // lds_addr
  g0[2] = (u32)ga;                              // global_addr[31:0]
  g0[3] = (u32)((ga >> 32) & 0x01FFFFFFu) | (2u << 30); // global_addr[56:32] | type=2

  v8i g1;
  g1[0] = (int)(2u << 16);      // workgroup_mask=0, data_size=2 (4 bytes)
  g1[1] = (int)(800u << 16);    // tensor_dim0[15:0]=800
  g1[2] = (int)(1u << 16);      // tensor_dim0 hi=0, tensor_dim1[15:0]=1
  g1[3] = (int)(800u << 16);    // tensor_dim1 hi=0, tile_dim0=800
  g1[4] = 1;                    // tile_dim1=1, tile_dim2=0
  g1[5] = 800;                  // tensor_dim0_stride lo
  g1[6] = (int)(800u << 16);    // stride0 hi=0, tensor_dim1_stride[15:0]=800
  g1[7] = 0;                    // tensor_dim1_stride hi

  v4i z4 = {0,0,0,0};
  v8i z8 = {0,0,0,0,0,0,0,0};
  __builtin_amdgcn_tensor_load_to_lds(g0, g1, z4, z4, z8, 0);
}

// Order-preserving float -> uint key (monotone bijection), for atomic max + exact equality.
__device__ __forceinline__ u32 fkey(float f) {
  int i = __float_as_int(f);
  return (i >= 0) ? ((u32)i | 0x80000000u) : ~((u32)i);
}

// IoU exactly as reference; contraction disabled so both passes produce identical bits.
__device__ __forceinline__ float iou_f(float ax1, float ay1, float ax2, float ay2,
                                       float gx1, float gy1, float gx2, float gy2)
{
  #pragma clang fp contract(off)
  float iw = fminf(ax2, gx2) - fmaxf(ax1, gx1) + 1.0f;
  float ih = fminf(ay2, gy2) - fmaxf(ay1, gy1) + 1.0f;
  iw = fmaxf(iw, 0.0f);
  ih = fmaxf(ih, 0.0f);
  float inter = iw * ih;
  float aa = (ax2 - ax1 + 1.0f) * (ay2 - ay1 + 1.0f);
  float ag = (gx2 - gx1 + 1.0f) * (gy2 - gy1 + 1.0f);
  return inter / (aa + ag - inter);
}

__device__ __forceinline__ void anchor_coords(int i, float& x1, float& y1, float& x2, float& y2)
{
  int a  = i % AA;
  int k  = i / AA;
  int xg = k % WW;
  int yg = k / WW;
  float sx = (float)(xg * ASTRIDE);
  float sy = (float)(yg * ASTRIDE);
  x1 = sx + BA[a*4+0];
  y1 = sy + BA[a*4+1];
  x2 = sx + BA[a*4+2];
  y2 = sy + BA[a*4+3];
}

__global__ void anchor_init_keys(u32* __restrict__ gkey)
{
  int t = threadIdx.x;
  if (t < NGT) gkey[t] = KEY_NEG1;
}

// Pass 1: per-gt column max of ov_in (outside anchors contribute -1 == init value).
__global__ void anchor_colmax(const float* __restrict__ gt,
                              const float* __restrict__ im_info,
                              u32* __restrict__ gkey)
{
  __shared__ float sGT[NGT*4];
  __shared__ float sX1[TPB], sY1[TPB], sX2[TPB], sY2[TPB];
  __shared__ int   sIn[TPB];

  int tid = threadIdx.x;
  if (tid < 32) {
    tdm_load_gt(gt, sGT);
    __builtin_amdgcn_s_wait_tensorcnt(0);
  }

  float imH = im_info[0];
  float imW = im_info[1];

  int i = blockIdx.x * TPB + tid;
  float x1, y1, x2, y2;
  anchor_coords(i, x1, y1, x2, y2);
  bool inside = (x1 >= 0.0f) && (y1 >= 0.0f) && (x2 < imW) && (y2 < imH);
  sX1[tid] = x1; sY1[tid] = y1; sX2[tid] = x2; sY2[tid] = y2;
  sIn[tid] = inside ? 1 : 0;
  __syncthreads();

  if (tid < NGT) {
    float gx1 = sGT[tid*4+0], gy1 = sGT[tid*4+1];
    float gx2 = sGT[tid*4+2], gy2 = sGT[tid*4+3];
    u32 mk = KEY_NEG1;
    for (int a = 0; a < TPB; ++a) {
      if (sIn[a]) {
        float ov = iou_f(sX1[a], sY1[a], sX2[a], sY2[a], gx1, gy1, gx2, gy2);
        u32 kk = fkey(ov);
        mk = (kk > mk) ? kk : mk;
      }
    }
    atomicMax(&gkey[tid], mk);
  }
}

// Pass 2: per-anchor labels + bbox targets.
__global__ void anchor_final(const float* __restrict__ gt,
                             const float* __restrict__ im_info,
                             const u32* __restrict__ gkey,
                             float* __restrict__ out)
{
  __shared__ float sGT[NGT*4];
  __shared__ u32   sKey[NGT];

  int tid = threadIdx.x;
  if (tid < 32) {
    tdm_load_gt(gt, sGT);
    __builtin_amdgcn_s_wait_tensorcnt(0);
  }
  if (tid < NGT) sKey[tid] = gkey[tid];
  __syncthreads();

  float imH = im_info[0];
  float imW = im_info[1];

  int i = blockIdx.x * TPB + tid;
  float x1, y1, x2, y2;
  anchor_coords(i, x1, y1, x2, y2);
  bool inside = (x1 >= 0.0f) && (y1 >= 0.0f) && (x2 < imW) && (y2 < imH);

  float best = -1.0f;
  int   arg  = 0;
  bool  gt_best = false;
  for (int j = 0; j < NGT; ++j) {
    float ov = iou_f(x1, y1, x2, y2,
                     sGT[j*4+0], sGT[j*4+1], sGT[j*4+2], sGT[j*4+3]);
    if (ov > best) { best = ov; arg = j; }        // first-occurrence argmax
    if (inside && (fkey(ov) == sKey[j])) gt_best = true;
  }

  float lab = -1.0f;
  if (inside && (best < 0.3f)) lab = 0.0f;                  // NEG_OVERLAP
  if (gt_best || (inside && (best >= 0.7f))) lab = 1.0f;    // POS_OVERLAP / gt-best
  out[i] = lab;

  float tx = 0.0f, ty = 0.0f, tw = 0.0f, th = 0.0f;
  if (inside) {
    #pragma clang fp contract(off)
    float gx1 = sGT[arg*4+0], gy1 = sGT[arg*4+1];
    float gx2 = sGT[arg*4+2], gy2 = sGT[arg*4+3];
    float aw = x2 - x1 + 1.0f, ah = y2 - y1 + 1.0f;
    float ax = x1 + 0.5f * aw, ay = y1 + 0.5f * ah;
    float gw = gx2 - gx1 + 1.0f, gh = gy2 - gy1 + 1.0f;
    float gx = gx1 + 0.5f * gw, gy = gy1 + 0.5f * gh;
    tx = (gx - ax) / aw;
    ty = (gy - ay) / ah;
    tw = logf(gw / aw);
    th = logf(gh / ah);
  }
  float* bt = out + NANCH;
  bt[i*4+0] = tx;
  bt[i*4+1] = ty;
  bt[i*4+2] = tw;
  bt[i*4+3] = th;
}

extern "C" void kernel_launch(void* const* d_in, const int* in_sizes, int n_in,
                              void* d_out, int out_size, void* d_ws, size_t ws_size,
                              hipStream_t stream)
{
  // d_in[0]=scores (unused: shape is compile-time), d_in[1]=im_info, d_in[2]=gt_boxes
  const float* im_info = (const float*)d_in[1];
  const float* gt      = (const float*)d_in[2];
  float*       out     = (float*)d_out;
  u32*         gkey    = (u32*)d_ws;   // 200 * 4 bytes of scratch

  anchor_init_keys<<<1, TPB, 0, stream>>>(gkey);
  anchor_colmax  <<<NBLK, TPB, 0, stream>>>(gt, im_info, gkey);
  anchor_final   <<<NBLK, TPB, 0, stream>>>(gt, im_info, gkey, out);
}